// Cubic_B_spline_FFD_3D_6511170421134
// MI455X (gfx1250) — compile-verified
//
#include <hip/hip_runtime.h>

typedef __attribute__((ext_vector_type(2))) float v2f;
typedef __attribute__((ext_vector_type(8))) float v8f;

#define IN_X 44
#define IN_Y 52
#define IN_Z 44
#define OUT_X 160
#define OUT_Y 192
#define OUT_Z 160
#define IN_SXY (IN_X * IN_Y)        // 2288
#define OUT_SXY (OUT_X * OUT_Y)     // 30720
#define IN_CH (IN_X * IN_Y * IN_Z)  // 100672
#define OUT_CH (OUT_X * OUT_Y * OUT_Z)

// LDS strides (padded to kill bank conflicts / divisions)
#define INL_ZS 64   // InL[z][y(8)][x(8)]
#define T1_ZS 57    // T1[zf][y(8)][x] -> odd stride, conflict-free lane reads
#define T2_ZS 144   // T2[zf][yf][x]
#define T2_YS 9

__device__ __forceinline__ float bspline(float t) {
  if (t < 1.f) return 2.f / 3.f + (0.5f * t - 1.f) * t * t;
  if (t < 2.f) { float s = 2.f - t; return s * s * s * (1.f / 6.f); }
  return 0.f;
}

__device__ __forceinline__ v8f wmma_f32(v2f a, v2f b, v8f c) {
  return __builtin_amdgcn_wmma_f32_16x16x4_f32(
      /*neg_a=*/false, a, /*neg_b=*/false, b,
      /*c_mod=*/(short)0, c, /*reuse_a=*/false, /*reuse_b=*/false);
}

// One workgroup (4 waves) computes a 16x16x16 output tile for one (n,c) from a
// 7x7x7 input tile, expanding z, then y, then x with
// Y[16,N] = W[16x8(K)] * X[8(K),N] via V_WMMA_F32_16X16X4_F32 (K looped 2x).
__global__ __launch_bounds__(128) void ffd_bspline_kernel(
    const float* __restrict__ vin, float* __restrict__ out) {
  __shared__ float Wl[16 * 8];          // B-spline weight matrix (col 7 == 0)
  __shared__ float InL[7 * INL_ZS];     // zero-padded 7x8x8 input tile
  __shared__ float T1[16 * T1_ZS];      // [zf][y][x]
  __shared__ float T2[16 * T2_ZS];      // [zf][yf][x]

  const int t = threadIdx.x;
  const int lane = t & 31;
  const int wv = t >> 5;     // wave id 0..3
  const int h = lane >> 4;   // lane half
  const int mn = lane & 15;  // A-row / B-col / D-col lane index

  int b = blockIdx.x;
  const int xt = b % 10; b /= 10;
  const int yt = b % 12; b /= 12;
  const int zt = b % 10; b /= 10;
  const int nc = b;  // fused n*3+c, 0..11

  // W[16][8]: W[i][j] = B(|(j - (i>>2)) - 1 - 0.25*(i&3)|); 128 elems, 1/thread.
  {
    const int i = t >> 3, j = t & 7;
    const int q = i >> 2, r = i & 3;
    const float u = (float)(j - q) - 1.f - 0.25f * (float)r;
    Wl[i * 8 + j] = bspline(fabsf(u));
  }

  // Load 7x7x7 input tile into zero-padded 7x8x8 LDS (all global accesses
  // provably in-bounds for these dims; pads written as 0 in the same loop).
  {
    const size_t ibase = (size_t)nc * IN_CH + (size_t)(4 * zt) * IN_SXY +
                         (size_t)(4 * yt) * IN_X + (size_t)(4 * xt);
    for (int e = t; e < 7 * INL_ZS; e += 128) {
      const int iz = e >> 6;
      const int iy = (e >> 3) & 7;
      const int ix = e & 7;
      float val = 0.f;
      if (iy < 7 && ix < 7)
        val = vin[ibase + (size_t)iz * IN_SXY + (size_t)iy * IN_X + (size_t)ix];
      InL[e] = val;
    }
  }
  __syncthreads();

  // A operand (weights), identical for all stages/chunks.
  // A layout (16x4 f32): row m = lane&15; VGPR0 = K {0|2}, VGPR1 = K {1|3}
  // split across lane halves; second WMMA covers K 4..7.
  v2f a0, a1;
  a0.x = Wl[mn * 8 + 2 * h];
  a0.y = Wl[mn * 8 + 2 * h + 1];
  a1.x = Wl[mn * 8 + 4 + 2 * h];
  a1.y = Wl[mn * 8 + 5 + 2 * h];
  const int k0 = 2 * h, k1 = 2 * h + 1;
  const int k2 = 4 + 2 * h;
  const int k3 = (5 + 2 * h > 6) ? 6 : (5 + 2 * h);  // K=7 weight is 0 -> clamp read

  // ---- Stage 1: expand z (7 -> 16). Columns nn: y = nn>>3, x = nn&7
  // (49 valid of 64); LDS offset is nn itself -> no div/mod anywhere.
  {
    const int nn = wv * 16 + mn;  // one chunk per wave
    v2f b0, b1;
    b0.x = InL[k0 * INL_ZS + nn]; b0.y = InL[k1 * INL_ZS + nn];
    b1.x = InL[k2 * INL_ZS + nn]; b1.y = InL[k3 * INL_ZS + nn];
    v8f c = {};
    c = wmma_f32(a0, b0, c);
    c = wmma_f32(a1, b1, c);
    if (nn < 56 && (nn & 7) < 7) {  // y<7 && x<7 (store-only lane mask)
#pragma unroll
      for (int v = 0; v < 8; ++v) T1[(v + 8 * h) * T1_ZS + nn] = c[v];
    }
  }
  __syncthreads();

  // ---- Stage 2: expand y (7 -> 16). Columns = (zf = lane, x = chunk).
  // Guard via readfirstlane -> scalar s_cbranch whole-wave skip, so no WMMA
  // ever executes with EXEC==0 (ISA: WMMA is not EXEC-skipped and requires
  // EXEC all-1s).
  {
    const int tb = mn * T1_ZS;
#pragma unroll
    for (int ii = 0; ii < 2; ++ii) {
      const int cc = wv * 2 + ii;  // x-coarse 0..6 (cc==7 skipped per-wave)
      if (__builtin_amdgcn_readfirstlane(cc) < 7) {
        v2f b0, b1;
        b0.x = T1[tb + k0 * 8 + cc]; b0.y = T1[tb + k1 * 8 + cc];
        b1.x = T1[tb + k2 * 8 + cc]; b1.y = T1[tb + k3 * 8 + cc];
        v8f c = {};
        c = wmma_f32(a0, b0, c);
        c = wmma_f32(a1, b1, c);
#pragma unroll
        for (int v = 0; v < 8; ++v)
          T2[mn * T2_ZS + (v + 8 * h) * T2_YS + cc] = c[v];
      }
    }
  }
  __syncthreads();

  // ---- Stage 3: expand x (7 -> 16). Columns = (zf = chunk, yf = lane).
  // D rows are x-fine = v + 8h -> 8 contiguous floats/lane -> 2x b128 stores.
  const size_t obase = (size_t)nc * OUT_CH + (size_t)(16 * zt) * OUT_SXY +
                       (size_t)(16 * yt) * OUT_X + (size_t)(16 * xt);
  float* op0 = out + obase + (size_t)mn * OUT_X + 8 * h;
#pragma unroll
  for (int jj = 0; jj < 4; ++jj) {
    const int cc = wv + 4 * jj;  // z-fine plane
    const int tb = cc * T2_ZS + mn * T2_YS;
    v2f b0, b1;
    b0.x = T2[tb + k0]; b0.y = T2[tb + k1];
    b1.x = T2[tb + k2]; b1.y = T2[tb + k3];
    v8f c = {};
    c = wmma_f32(a0, b0, c);
    c = wmma_f32(a1, b1, c);
    float* op = op0 + (size_t)cc * OUT_SXY;
    float4 lo = make_float4(c[0], c[1], c[2], c[3]);
    float4 hi = make_float4(c[4], c[5], c[6], c[7]);
    *reinterpret_cast<float4*>(op) = lo;
    *reinterpret_cast<float4*>(op + 4) = hi;
  }
}

extern "C" void kernel_launch(void* const* d_in, const int* in_sizes, int n_in,
                              void* d_out, int out_size, void* d_ws, size_t ws_size,
                              hipStream_t stream) {
  (void)in_sizes; (void)n_in; (void)out_size; (void)d_ws; (void)ws_size;
  const float* v = (const float*)d_in[0];
  float* out = (float*)d_out;
  // tiles: 10 (x) * 12 (y) * 10 (z) * 12 (n*c) = 14400 workgroups, 4 waves each
  ffd_bspline_kernel<<<dim3(14400), dim3(128), 0, stream>>>(v, out);
}